// GraphConvNet_38259568672943
// MI455X (gfx1250) — compile-verified
//
#include <hip/hip_runtime.h>
#include <hip/hip_bf16.h>
#include <stdint.h>

#define N_ATOMS   100000
#define N_BONDS   200000
#define MAX_NB    6
#define ATOM_F    39
#define BOND_IN   50        // 39 + 11
#define HID       128
#define DEPTH     6
#define W_O_K     167       // 39 + 128
#define WSTRIDE   132       // 128 + 4 DWORD pad (TDM pad feature), bank-conflict free

typedef __attribute__((ext_vector_type(2))) float v2f;
typedef __attribute__((ext_vector_type(8))) float v8f;
typedef __attribute__((ext_vector_type(4))) unsigned int u32x4;
typedef __attribute__((ext_vector_type(8))) int i32x8;
typedef __attribute__((ext_vector_type(4))) int i32x4;

// ---------------------------------------------------------------------------
// Kernel 1: binput = fbonds @ W_i.T      [N_BONDS, 128], K = 50 (padded to 52)
// One block = 16 bond rows; 8 waves each own one 16x16 N-tile.
// ---------------------------------------------------------------------------
__global__ __launch_bounds__(256) void k_bond_input(
    const float* __restrict__ fbonds, const float* __restrict__ W_i,
    float* __restrict__ binput)
{
    __shared__ float sA[16][56];              // 16 x 52 used, cols 50/51 zero
    const int tile = blockIdx.x;
    const int row0 = tile * 16;
    const int t = threadIdx.x;

    for (int i = t; i < 16 * 52; i += 256) {
        int r = i / 52, c = i % 52;
        sA[r][c] = (c < BOND_IN) ? fbonds[(row0 + r) * BOND_IN + c] : 0.0f;
    }
    __syncthreads();

    const int wave = t >> 5, lane = t & 31;
    const int lrow = lane & 15, hi = lane >> 4;
    const int n = wave * 16 + lrow;

    v8f acc = {};
    #pragma unroll
    for (int k0 = 0; k0 < 52; k0 += 4) {
        const int ka = k0 + hi * 2;
        // A columns >= 50 are zero in LDS, so B just needs an in-bounds
        // address (clamped), no value guard -> no EXEC-divergent loads.
        const int kb = (ka < BOND_IN - 1) ? ka : 0;
        v2f a; a.x = sA[lrow][ka]; a.y = sA[lrow][ka + 1];
        v2f b; b.x = W_i[n * BOND_IN + kb]; b.y = W_i[n * BOND_IN + kb + 1];
        acc = __builtin_amdgcn_wmma_f32_16x16x4_f32(
            false, a, false, b, (short)0, acc, false, false);
    }
    #pragma unroll
    for (int r = 0; r < 8; ++r) {
        const int m = row0 + r + hi * 8;
        binput[m * HID + n] = acc[r];
    }
}

// ---------------------------------------------------------------------------
// Kernel 2: one message-passing round (fused gather + GEMM + residual).
//   nei   = sum_j relu(Xin[bgraph[b][j]])      (ReLU applied at gather time)
//   Xout  = binput + nei @ W_h.T               (pre-activation stored)
// W_h (64KB) is staged into LDS by the Tensor Data Mover with a 4-DWORD pad
// every 128 DWORDs (row stride 132 -> conflict-free B-fragment reads),
// overlapped with the cooperative neighbor gather.
// ---------------------------------------------------------------------------
__global__ __launch_bounds__(256) void k_mp_iter(
    const float* __restrict__ Xin, const int* __restrict__ bgraph,
    const float* __restrict__ W_h, const float* __restrict__ binput,
    float* __restrict__ Xout)
{
    __shared__ float sW[HID * WSTRIDE];       // W_h, row stride 132
    __shared__ float sN[16][WSTRIDE];         // gathered nei sums
    const int tile = blockIdx.x;
    const int row0 = tile * 16;
    const int t = threadIdx.x;
    const int wave = t >> 5;

    // --- issue async TDM copy of W_h -> LDS (wave 0 only; EXEC ignored) ---
    if (wave == 0) {
        const uint32_t lds = (uint32_t)(uintptr_t)(void*)sW;   // LDS aperture: low 32 bits = LDS offset
        const uint64_t ga  = (uint64_t)(uintptr_t)W_h;
        u32x4 g0;
        g0.x = 1u;                                             // count = 1 valid descriptor
        g0.y = lds;                                            // lds_addr
        g0.z = (uint32_t)ga;                                   // global_addr[31:0]
        g0.w = (uint32_t)((ga >> 32) & 0x1FFFFFFu) | (2u << 30); // global_addr[56:32] | type=2
        i32x8 g1;
        g1[0] = (2 << 16)                                      // data_size = 4B
              | (1 << 20)                                      // pad_enable
              | (6 << 22)                                      // pad_interval: every 128 DWORDs
              | (3 << 25);                                     // pad_amount: 4 DWORDs
        g1[1] = (HID << 16);                                   // tensor_dim0 = 128 (lo16)
        g1[2] = (HID << 16);                                   // tensor_dim1 = 128 (lo16)
        g1[3] = (HID << 16);                                   // tile_dim0 = 128
        g1[4] = HID;                                           // tile_dim1 = 128, tile_dim2 = 0
        g1[5] = HID;                                           // tensor_dim0_stride = 128
        g1[6] = 0;
        g1[7] = 0;
        const i32x4 z4 = {0, 0, 0, 0};                         // groups 2/3 unused (2D tensor)
        const i32x8 z8 = {0, 0, 0, 0, 0, 0, 0, 0};
        __builtin_amdgcn_tensor_load_to_lds(g0, g1, z4, z4, z8, 0);
    }

    // --- cooperative gather (overlaps the TDM): 16 rows x 32 float4 cols ---
    for (int u = t; u < 16 * 32; u += 256) {
        const int r = u >> 5;
        const int c4 = (u & 31) * 4;
        const int bond = row0 + r;
        float4 s = make_float4(0.f, 0.f, 0.f, 0.f);
        #pragma unroll
        for (int j = 0; j < MAX_NB; ++j) {
            const int src = bgraph[bond * MAX_NB + j];
            const float4 v = *(const float4*)(Xin + (size_t)src * HID + c4);
            s.x += fmaxf(v.x, 0.f); s.y += fmaxf(v.y, 0.f);
            s.z += fmaxf(v.z, 0.f); s.w += fmaxf(v.w, 0.f);
        }
        sN[r][c4 + 0] = s.x; sN[r][c4 + 1] = s.y;
        sN[r][c4 + 2] = s.z; sN[r][c4 + 3] = s.w;
    }

    if (wave == 0) __builtin_amdgcn_s_wait_tensorcnt(0);
    __syncthreads();

    const int lane = t & 31;
    const int lrow = lane & 15, hi = lane >> 4;
    const int n = wave * 16 + lrow;

    v8f acc = {};
    #pragma unroll 8
    for (int k0 = 0; k0 < HID; k0 += 4) {
        const int ka = k0 + hi * 2;
        v2f a; a.x = sN[lrow][ka];           a.y = sN[lrow][ka + 1];
        v2f b; b.x = sW[n * WSTRIDE + ka];   b.y = sW[n * WSTRIDE + ka + 1];
        acc = __builtin_amdgcn_wmma_f32_16x16x4_f32(
            false, a, false, b, (short)0, acc, false, false);
    }
    #pragma unroll
    for (int r = 0; r < 8; ++r) {
        const int m = row0 + r + hi * 8;
        const int idx = m * HID + n;
        Xout[idx] = binput[idx] + acc[r];     // relu deferred to next gather
    }
}

// ---------------------------------------------------------------------------
// Kernel 3: atom readout.
//   a_nei  = sum_j relu(Xfin[agraph[a][j]])
//   out    = relu(concat(fatoms, a_nei) @ W_o.T + b_o)
// K = 167 handled as 40 (fatoms, col 39 zero-padded) + 128 (a_nei).
// ---------------------------------------------------------------------------
__global__ __launch_bounds__(256) void k_atom_out(
    const float* __restrict__ Xfin, const int* __restrict__ agraph,
    const float* __restrict__ fatoms, const float* __restrict__ W_o,
    const float* __restrict__ b_o, float* __restrict__ out)
{
    __shared__ float sN[16][WSTRIDE];
    __shared__ float sF[16][40];
    const int tile = blockIdx.x;
    const int row0 = tile * 16;
    const int t = threadIdx.x;

    for (int u = t; u < 16 * 32; u += 256) {
        const int r = u >> 5;
        const int c4 = (u & 31) * 4;
        const int atom = row0 + r;
        float4 s = make_float4(0.f, 0.f, 0.f, 0.f);
        #pragma unroll
        for (int j = 0; j < MAX_NB; ++j) {
            const int src = agraph[atom * MAX_NB + j];
            const float4 v = *(const float4*)(Xfin + (size_t)src * HID + c4);
            s.x += fmaxf(v.x, 0.f); s.y += fmaxf(v.y, 0.f);
            s.z += fmaxf(v.z, 0.f); s.w += fmaxf(v.w, 0.f);
        }
        sN[r][c4 + 0] = s.x; sN[r][c4 + 1] = s.y;
        sN[r][c4 + 2] = s.z; sN[r][c4 + 3] = s.w;
    }
    for (int i = t; i < 16 * 40; i += 256) {
        int r = i / 40, c = i % 40;
        sF[r][c] = (c < ATOM_F) ? fatoms[(row0 + r) * ATOM_F + c] : 0.0f;
    }
    const int wave = t >> 5, lane = t & 31;
    const int lrow = lane & 15, hi = lane >> 4;
    const int n = wave * 16 + lrow;
    __builtin_prefetch(&W_o[n * W_O_K], 0, 3);   // global_prefetch_b8 (L2-resident)
    __syncthreads();

    v8f acc = {};
    // fatoms block: K 0..39 (col 39 of sF is zero => W_o[:,39] contributes 0 here)
    #pragma unroll
    for (int k0 = 0; k0 < 40; k0 += 4) {
        const int ka = k0 + hi * 2;
        v2f a; a.x = sF[lrow][ka]; a.y = sF[lrow][ka + 1];
        v2f b; b.x = W_o[n * W_O_K + ka]; b.y = W_o[n * W_O_K + ka + 1];
        acc = __builtin_amdgcn_wmma_f32_16x16x4_f32(
            false, a, false, b, (short)0, acc, false, false);
    }
    // a_nei block: K 39..166
    #pragma unroll 8
    for (int k0 = 0; k0 < HID; k0 += 4) {
        const int ka = k0 + hi * 2;
        v2f a; a.x = sN[lrow][ka]; a.y = sN[lrow][ka + 1];
        v2f b; b.x = W_o[n * W_O_K + ATOM_F + ka];
               b.y = W_o[n * W_O_K + ATOM_F + ka + 1];
        acc = __builtin_amdgcn_wmma_f32_16x16x4_f32(
            false, a, false, b, (short)0, acc, false, false);
    }
    const float bias = b_o[n];
    #pragma unroll
    for (int r = 0; r < 8; ++r) {
        const int m = row0 + r + hi * 8;
        out[m * HID + n] = fmaxf(acc[r] + bias, 0.0f);
    }
}

// ---------------------------------------------------------------------------
extern "C" void kernel_launch(void* const* d_in, const int* in_sizes, int n_in,
                              void* d_out, int out_size, void* d_ws, size_t ws_size,
                              hipStream_t stream) {
    const float* fatoms = (const float*)d_in[0];
    const float* fbonds = (const float*)d_in[1];
    const int*   agraph = (const int*)d_in[2];
    const int*   bgraph = (const int*)d_in[3];
    const float* W_i    = (const float*)d_in[4];
    const float* W_h    = (const float*)d_in[5];
    const float* W_o    = (const float*)d_in[6];
    const float* b_o    = (const float*)d_in[7];
    float* out = (float*)d_out;

    const size_t bufElems = (size_t)N_BONDS * HID;
    float* binput = (float*)d_ws;
    float* Xa = binput + bufElems;
    float* Xb = Xa + bufElems;

    k_bond_input<<<N_BONDS / 16, 256, 0, stream>>>(fbonds, W_i, binput);

    const float* cur = binput;
    float* pp[2] = {Xa, Xb};
    for (int d = 0; d < DEPTH - 1; ++d) {
        float* nxt = pp[d & 1];
        k_mp_iter<<<N_BONDS / 16, 256, 0, stream>>>(cur, bgraph, W_h, binput, nxt);
        cur = nxt;
    }

    k_atom_out<<<N_ATOMS / 16, 256, 0, stream>>>(cur, agraph, fatoms, W_o, b_o, out);
}